// model0_0_72988674228292
// MI455X (gfx1250) — compile-verified
//
#include <hip/hip_runtime.h>

typedef __attribute__((ext_vector_type(2))) float v2f;
typedef __attribute__((ext_vector_type(8))) float v8f;

// D = A(16x4) * B(4x16) + C, fp32 WMMA (CDNA5 wave32)
#define WMMA4(A, B, C) \
  __builtin_amdgcn_wmma_f32_16x16x4_f32(false, (A), false, (B), (short)0, (C), false, false)

__global__ __launch_bounds__(32)
void point_transformer_group_kernel(
    const float* __restrict__ data,
    const float* __restrict__ w_qkv,
    const float* __restrict__ pos_w1, const float* __restrict__ pos_b1,
    const float* __restrict__ pos_w2, const float* __restrict__ pos_b2,
    const float* __restrict__ attn_w1, const float* __restrict__ attn_b1,
    const float* __restrict__ attn_w2, const float* __restrict__ attn_b2,
    const float* __restrict__ mlp_w1, const float* __restrict__ mlp_b1,
    const float* __restrict__ mlp_w2, const float* __restrict__ mlp_b2,
    const float* __restrict__ mlp_w3, const float* __restrict__ mlp_b3,
    float* __restrict__ out)
{
  __shared__ float s_pos[16][3];
  __shared__ float s_q[16][3];
  __shared__ float s_k[16][3];
  __shared__ float s_v[16][3];
  __shared__ float s_h1[16][64];   // pos-MLP hidden tile (16 pairs x 64)
  __shared__ float s_hA[16][16];   // attn-MLP hidden tile (16 pairs x 12, zero-padded)
  __shared__ float s_re[16][16];   // rel_emb tile (cols 3..15 are exact zeros)
  __shared__ float s_sim[16][16];  // sim tile (cols 3..15 are exact zeros)
  __shared__ float s_e[16][3];
  __shared__ float s_sa[48];
  __shared__ float s_fc[48];

  const int lane = threadIdx.x;
  const int lo = lane & 15;
  const int hi = lane >> 4;        // 0: K=0/1 half, 1: K=2/3 half
  const int c0 = hi * 2;           // xyz component handled by this half for K slot 0
  const int g = blockIdx.x;
  const float* xg = data + (size_t)g * 48;

  // ---- load x; lanes 0..15 compute q,k,v for their neighbor ----
  if (lane < 16) {
    float x0 = xg[lane * 3 + 0], x1 = xg[lane * 3 + 1], x2 = xg[lane * 3 + 2];
    s_pos[lane][0] = x0; s_pos[lane][1] = x1; s_pos[lane][2] = x2;
    #pragma unroll
    for (int o = 0; o < 3; ++o) {
      s_q[lane][o] = x0 * w_qkv[0 * 9 + o]     + x1 * w_qkv[1 * 9 + o]     + x2 * w_qkv[2 * 9 + o];
      s_k[lane][o] = x0 * w_qkv[0 * 9 + 3 + o] + x1 * w_qkv[1 * 9 + 3 + o] + x2 * w_qkv[2 * 9 + 3 + o];
      s_v[lane][o] = x0 * w_qkv[0 * 9 + 6 + o] + x1 * w_qkv[1 * 9 + 6 + o] + x2 * w_qkv[2 * 9 + 6 + o];
    }
  }

  // ---- preload WMMA B operands (weights, uniform across blocks -> L2 hits) ----
  // pos layer1 B: [4x64] in 4 N-tiles of 16 cols. Row K=3 is zero pad.
  v2f bp1[4]; float cb1[4];
  #pragma unroll
  for (int nt = 0; nt < 4; ++nt) {
    int c = nt * 16 + lo;
    bp1[nt].x = hi ? pos_w1[2 * 64 + c] : pos_w1[0 * 64 + c];
    bp1[nt].y = hi ? 0.0f               : pos_w1[1 * 64 + c];
    cb1[nt]   = pos_b1[c];
  }
  // pos layer2 B: [64x3] in 16 K-chunks of 4; cols >=3 are zero.
  v2f bp2[16];
  float cb2 = (lo < 3) ? pos_b2[lo] : 0.0f;
  #pragma unroll
  for (int kc = 0; kc < 16; ++kc) {
    int k0 = kc * 4 + c0;
    bp2[kc].x = (lo < 3) ? pos_w2[k0 * 3 + lo]       : 0.0f;
    bp2[kc].y = (lo < 3) ? pos_w2[(k0 + 1) * 3 + lo] : 0.0f;
  }
  // attn layer1 B: [4x12] (cols 12..15 zero, row K=3 zero)
  v2f ba1;
  ba1.x = (lo < 12) ? (hi ? attn_w1[2 * 12 + lo] : attn_w1[0 * 12 + lo]) : 0.0f;
  ba1.y = (lo < 12 && !hi) ? attn_w1[1 * 12 + lo] : 0.0f;
  float cba1 = (lo < 12) ? attn_b1[lo] : 0.0f;
  // attn layer2 B: [12x3] padded to K=16 -> 4 chunks
  v2f ba2[4];
  float cba2 = (lo < 3) ? attn_b2[lo] : 0.0f;
  #pragma unroll
  for (int kc = 0; kc < 4; ++kc) {
    int k0 = kc * 4 + c0;
    ba2[kc].x = (k0 < 12 && lo < 3)     ? attn_w2[k0 * 3 + lo]       : 0.0f;
    ba2[kc].y = (k0 + 1 < 12 && lo < 3) ? attn_w2[(k0 + 1) * 3 + lo] : 0.0f;
  }

  __syncthreads();

  // ---- tile loop: i = t fixed, rows m = j = 0..15 ----
  for (int t = 0; t < 16; ++t) {
    // A = rel_pos[t, j] rows (K = xyz padded to 4), fully branchless:
    // lane half 'hi' owns K slots {2*hi, 2*hi+1}; slot 3 is the zero pad.
    float pi_c = s_pos[t][c0];
    float pi_y = s_pos[t][1];
    v2f a;
    a.x = pi_c - s_pos[lo][c0];
    a.y = hi ? 0.0f : (pi_y - s_pos[lo][1]);

    // pos-MLP layer1: [16x4] @ [4x64] as 4 N-tiles, relu, stash to LDS
    #pragma unroll
    for (int nt = 0; nt < 4; ++nt) {
      v8f c;
      #pragma unroll
      for (int r = 0; r < 8; ++r) c[r] = cb1[nt];
      v8f d = WMMA4(a, bp1[nt], c);
      #pragma unroll
      for (int r = 0; r < 8; ++r) {
        float h = d[r] > 0.0f ? d[r] : 0.0f;
        s_h1[r + 8 * hi][nt * 16 + lo] = h;
      }
    }
    __syncthreads();

    // pos-MLP layer2: [16x64] @ [64x3], 16 K-chunks accumulated in D
    v8f dre;
    #pragma unroll
    for (int r = 0; r < 8; ++r) dre[r] = cb2;
    #pragma unroll
    for (int kc = 0; kc < 16; ++kc) {
      v2f a2;
      a2.x = s_h1[lo][kc * 4 + c0 + 0];
      a2.y = s_h1[lo][kc * 4 + c0 + 1];
      dre = WMMA4(a2, bp2[kc], dre);
    }
    // cols >= 3 of dre are exactly 0 (zero B cols, zero bias) -> store all 16
    #pragma unroll
    for (int r = 0; r < 8; ++r) s_re[r + 8 * hi][lo] = dre[r];
    __syncthreads();

    // attn-MLP layer1: A = (q[t]-k[j]) + rel_emb[t,j], branchless as above
    float qi_c = s_q[t][c0];
    float qi_y = s_q[t][1];
    v2f aa;
    aa.x = qi_c - s_k[lo][c0] + s_re[lo][c0];
    aa.y = hi ? 0.0f : (qi_y - s_k[lo][1] + s_re[lo][1]);
    v8f cA;
    #pragma unroll
    for (int r = 0; r < 8; ++r) cA[r] = cba1;
    v8f dA = WMMA4(aa, ba1, cA);
    #pragma unroll
    for (int r = 0; r < 8; ++r) {
      float h = dA[r] > 0.0f ? dA[r] : 0.0f;
      s_hA[r + 8 * hi][lo] = h;  // cols 12..15 are exactly 0
    }
    __syncthreads();

    // attn-MLP layer2: K=12 padded to 16 -> 4 chunks
    v8f dS;
    #pragma unroll
    for (int r = 0; r < 8; ++r) dS[r] = cba2;
    #pragma unroll
    for (int kc = 0; kc < 4; ++kc) {
      v2f a2;
      a2.x = s_hA[lo][kc * 4 + c0 + 0];
      a2.y = s_hA[lo][kc * 4 + c0 + 1];
      dS = WMMA4(a2, ba2[kc], dS);
    }
    #pragma unroll
    for (int r = 0; r < 8; ++r) s_sim[r + 8 * hi][lo] = dS[r];
    __syncthreads();

    // softmax over j per channel d, then aggregate with v_ij = v[j] + rel_emb
    if (lane < 3) {
      float mx = -3.4e38f;
      for (int j = 0; j < 16; ++j) mx = fmaxf(mx, s_sim[j][lane]);
      float sum = 0.0f;
      for (int j = 0; j < 16; ++j) {
        float e = __expf(s_sim[j][lane] - mx);
        s_e[j][lane] = e;
        sum += e;
      }
      float inv = 1.0f / sum;
      float agg = 0.0f;
      for (int j = 0; j < 16; ++j)
        agg += s_e[j][lane] * inv * (s_v[j][lane] + s_re[j][lane]);
      s_sa[t * 3 + lane] = agg;
    }
    __syncthreads();
  }

  // ---- final MLP 48 -> 48 -> 48 -> 1 (plain VALU; ~9 KFLOP) ----
  {
    float acc1 = mlp_b1[lane];
    float acc2 = (lane < 16) ? mlp_b1[lane + 32] : 0.0f;
    for (int k = 0; k < 48; ++k) {
      float s = s_sa[k];
      acc1 += s * mlp_w1[k * 48 + lane];
      if (lane < 16) acc2 += s * mlp_w1[k * 48 + lane + 32];
    }
    __syncthreads();
    s_fc[lane] = acc1 > 0.0f ? acc1 : 0.0f;
    if (lane < 16) s_fc[lane + 32] = acc2 > 0.0f ? acc2 : 0.0f;
    __syncthreads();

    acc1 = mlp_b2[lane];
    acc2 = (lane < 16) ? mlp_b2[lane + 32] : 0.0f;
    for (int k = 0; k < 48; ++k) {
      float s = s_fc[k];
      acc1 += s * mlp_w2[k * 48 + lane];
      if (lane < 16) acc2 += s * mlp_w2[k * 48 + lane + 32];
    }
    __syncthreads();
    s_sa[lane] = acc1 > 0.0f ? acc1 : 0.0f;
    if (lane < 16) s_sa[lane + 32] = acc2 > 0.0f ? acc2 : 0.0f;
    __syncthreads();

    if (lane == 0) {
      float acc = mlp_b3[0];
      for (int k = 0; k < 48; ++k) acc += s_sa[k] * mlp_w3[k];
      out[g] = acc;
    }
  }
}

extern "C" void kernel_launch(void* const* d_in, const int* in_sizes, int n_in,
                              void* d_out, int out_size, void* d_ws, size_t ws_size,
                              hipStream_t stream) {
  // setup_inputs order:
  // 0 original_points (unused), 1 data, 2 w_qkv, 3 pos_w1, 4 pos_b1, 5 pos_w2,
  // 6 pos_b2, 7 attn_w1, 8 attn_b1, 9 attn_w2, 10 attn_b2, 11 mlp_w1, 12 mlp_b1,
  // 13 mlp_w2, 14 mlp_b2, 15 mlp_w3, 16 mlp_b3
  const float* data    = (const float*)d_in[1];
  const float* w_qkv   = (const float*)d_in[2];
  const float* pos_w1  = (const float*)d_in[3];
  const float* pos_b1  = (const float*)d_in[4];
  const float* pos_w2  = (const float*)d_in[5];
  const float* pos_b2  = (const float*)d_in[6];
  const float* attn_w1 = (const float*)d_in[7];
  const float* attn_b1 = (const float*)d_in[8];
  const float* attn_w2 = (const float*)d_in[9];
  const float* attn_b2 = (const float*)d_in[10];
  const float* mlp_w1  = (const float*)d_in[11];
  const float* mlp_b1  = (const float*)d_in[12];
  const float* mlp_w2  = (const float*)d_in[13];
  const float* mlp_b2  = (const float*)d_in[14];
  const float* mlp_w3  = (const float*)d_in[15];
  const float* mlp_b3  = (const float*)d_in[16];
  float* out = (float*)d_out;

  const int BG = in_sizes[1] / 48;  // B*GROUPS groups, one wave32 each
  point_transformer_group_kernel<<<BG, 32, 0, stream>>>(
      data, w_qkv, pos_w1, pos_b1, pos_w2, pos_b2,
      attn_w1, attn_b1, attn_w2, attn_b2,
      mlp_w1, mlp_b1, mlp_w2, mlp_b2, mlp_w3, mlp_b3, out);
}